// BiquadEq_16879221473514
// MI455X (gfx1250) — compile-verified
//
#include <hip/hip_runtime.h>
#include <hip/hip_bf16.h>

// Biquad peaking EQ over [B=32, C=2, T=524288] f32.
// Warm-up chunking (zero-state start, W=512, pole decay ~0.919^512 ~ 1e-19 -> fp32-exact)
// gives 16384 independent columns. Inner recurrence blocked by 16 steps:
//   Y(16x16) = G(16x20) x V(20x16)  via five chained V_WMMA_F32_16X16X4_F32 per wave.
// V slice0 ordering avoids any cross-lane shuffle between blocks:
//   K0=x[-2], K1=x[-1]  (lower lane half; contiguous float2 from memory, prefetched)
//   K2=y[-1], K3=y[-2]  (upper lane half; = acc[7],acc[6], produced in those very lanes)
// Loads are software-pipelined 2 steps deep + global_prefetch ~12 steps ahead.

typedef __attribute__((ext_vector_type(2))) float v2f;
typedef __attribute__((ext_vector_type(8))) float v8f;

#define SR            44100.0f
#define T_LEN         524288
#define NCH           64                      // 32 * 2
#define CHUNK         2048
#define WARM          512
#define CHUNKS_PER_CH (T_LEN / CHUNK)         // 256
#define NCOLS         (NCH * CHUNKS_PER_CH)   // 16384
#define NSTEPS        ((CHUNK + WARM) / 16)   // 160 (uniform trip count: EXEC all-1s for WMMA)
#define OUTSTEPS      (CHUNK / 16)            // 128
#define PF_STEPS      12                      // prefetch distance (steps of 16 floats)

__global__ __launch_bounds__(128) void biquad_wmma(const float* __restrict__ x,
                                                   const float* __restrict__ cfp,
                                                   const float* __restrict__ qp,
                                                   const float* __restrict__ gp,
                                                   float* __restrict__ out)
{
    // G[t][c]: response at block-time t to unit value in V-component c.
    // c0: x[-2], c1: x[-1], c2: y[-1], c3: y[-2], c4+j: x[j] (j=0..15).
    __shared__ float G[16][20];

    if (threadIdx.x == 0) {
        float fc = cfp[0], qv = qp[0], gn = gp[0];
        float g  = fminf(fmaxf(gn, 0.1f), 10.0f);
        float w0 = 2.0f * 3.14159265358979323846f * fc / SR;
        float Aa = __expf(g * 0.025f * __logf(10.0f));       // 10^(g/40)
        float al = __sinf(w0) / (2.0f * qv);
        float cw = __cosf(w0);
        float a0  = 1.0f + al / Aa;
        float b0c = (1.0f + al * Aa) / a0;
        float b1c = (-2.0f * cw) / a0;
        float b2c = (1.0f - al * Aa) / a0;
        float a1c = b1c;
        float a2c = (1.0f - al / Aa) / a0;

        for (int c = 0; c < 20; ++c) {
            float p1 = (c == 2) ? 1.0f : 0.0f;   // y[-1] impulse
            float p2 = (c == 3) ? 1.0f : 0.0f;   // y[-2] impulse
            for (int t = 0; t < 16; ++t) {
                float xn = (c >= 4 && t == (c - 4)) ? 1.0f : 0.0f;
                float x1 = ((c >= 4 && t == (c - 3)) || (c == 1 && t == 0)) ? 1.0f : 0.0f;
                float x2 = ((c >= 4 && t == (c - 2)) || (c == 1 && t == 1)
                            || (c == 0 && t == 0)) ? 1.0f : 0.0f;
                float v = b0c * xn + b1c * x1 + b2c * x2 - a1c * p1 - a2c * p2;
                G[t][c] = v;
                p2 = p1; p1 = v;
            }
        }
    }
    __syncthreads();

    const int lane = threadIdx.x & 31;
    const int hh   = lane >> 4;          // lane half: K subset {0,1} vs {2,3} per slice
    const int m    = lane & 15;          // A row (M) / B,C,D column (N)
    const int wave = (int)((blockIdx.x * blockDim.x + threadIdx.x) >> 5);
    const int col  = wave * 16 + m;      // this lane's column = (channel, chunk)
    const int ch   = col / CHUNKS_PER_CH;
    const int ck   = col % CHUNKS_PER_CH;
    const int t0   = ck * CHUNK;
    const int tstart = (ck == 0) ? 0 : (t0 - WARM);
    const int sstart = (ck == 0) ? 0 : (WARM / 16);

    // A-matrix slices: 16x4 f32 layout -> lane holds row M=m; v0:K={0|2}, v1:K={1|3} by half.
    v2f ga[5];
#pragma unroll
    for (int s = 0; s < 5; ++s) {
        ga[s].x = G[m][4 * s + 2 * hh + 0];
        ga[s].y = G[m][4 * s + 2 * hh + 1];
    }

    const float* xp = x   + (size_t)ch * T_LEN + tstart;
    float*       yp = out + (size_t)ch * T_LEN + tstart;

    // Software pipeline, depth 2: current (c*), next (n*), far (f*) buffer sets.
    const float* p0 = xp + 2 * hh;
    v2f cb1 = *(const v2f*)(p0 + 0);
    v2f cb2 = *(const v2f*)(p0 + 4);
    v2f cb3 = *(const v2f*)(p0 + 8);
    v2f cb4 = *(const v2f*)(p0 + 12);
    v2f cst; cst.x = 0.0f; cst.y = 0.0f;   // (x[-2], x[-1]) for lower half; zero at warm start

    const float* p1 = xp + 16 + 2 * hh;
    v2f nb1 = *(const v2f*)(p1 + 0);
    v2f nb2 = *(const v2f*)(p1 + 4);
    v2f nb3 = *(const v2f*)(p1 + 8);
    v2f nb4 = *(const v2f*)(p1 + 12);
    v2f nst = *(const v2f*)(xp + 16 - 2);  // step 1's (x[-2], x[-1])

    float y1s = 0.0f, y2s = 0.0f;          // (y[-1], y[-2]); live in upper-half lanes

    for (int step = 0; step < NSTEPS; ++step) {
        // Issue loads for step+2 (clamped in-bounds; results discarded on last steps).
        const int sfar = (step + 2 < NSTEPS) ? (step + 2) : (NSTEPS - 1);
        const float* pf = xp + sfar * 16 + 2 * hh;
        v2f fb1 = *(const v2f*)(pf + 0);
        v2f fb2 = *(const v2f*)(pf + 4);
        v2f fb3 = *(const v2f*)(pf + 8);
        v2f fb4 = *(const v2f*)(pf + 12);
        v2f fst = *(const v2f*)(xp + sfar * 16 - 2);   // sfar >= 2 -> always in-bounds

        // Warm the cache hierarchy well ahead of the demand stream (clamped in-bounds).
        const int spf = (step + PF_STEPS < NSTEPS) ? (step + PF_STEPS) : (NSTEPS - 1);
        __builtin_prefetch(xp + spf * 16 + 2 * hh, 0, 3);

        // Slice-0 B operand: lower half = (x[-2], x[-1]); upper half = (y[-1], y[-2]).
        v2f b0v;
        b0v.x = hh ? y1s : cst.x;
        b0v.y = hh ? y2s : cst.y;

        v8f acc = {};
        acc = __builtin_amdgcn_wmma_f32_16x16x4_f32(false, ga[0], false, b0v, (short)0, acc, false, false);
        acc = __builtin_amdgcn_wmma_f32_16x16x4_f32(false, ga[1], false, cb1, (short)0, acc, false, false);
        acc = __builtin_amdgcn_wmma_f32_16x16x4_f32(false, ga[2], false, cb2, (short)0, acc, false, false);
        acc = __builtin_amdgcn_wmma_f32_16x16x4_f32(false, ga[3], false, cb3, (short)0, acc, false, false);
        acc = __builtin_amdgcn_wmma_f32_16x16x4_f32(false, ga[4], false, cb4, (short)0, acc, false, false);

        // New y-state: Y rows 15/14 = acc[7]/acc[6], already resident in the upper-half
        // lanes that consume them next step (C/D layout: M = r + 8*hh). No shuffle.
        y1s = acc[7];
        y2s = acc[6];

        // Emit only the non-warm-up window [t0, t0+CHUNK).
        if (step >= sstart && step < sstart + OUTSTEPS) {
            // Lane holds Y[M = r + 8*hh][N = m] in acc[r] -> 8 time-contiguous floats.
            float* q0 = yp + step * 16 + 8 * hh;        // 16B-aligned
            float4 lo = make_float4(acc[0], acc[1], acc[2], acc[3]);
            float4 hi = make_float4(acc[4], acc[5], acc[6], acc[7]);
            *(float4*)(q0 + 0) = lo;
            *(float4*)(q0 + 4) = hi;
        }

        // Rotate pipeline buffers.
        cb1 = nb1; cb2 = nb2; cb3 = nb3; cb4 = nb4; cst = nst;
        nb1 = fb1; nb2 = fb2; nb3 = fb3; nb4 = fb4; nst = fst;
    }
}

extern "C" void kernel_launch(void* const* d_in, const int* in_sizes, int n_in,
                              void* d_out, int out_size, void* d_ws, size_t ws_size,
                              hipStream_t stream) {
    const float* x  = (const float*)d_in[0];
    const float* cf = (const float*)d_in[1];
    const float* qv = (const float*)d_in[2];
    const float* gn = (const float*)d_in[3];
    float* out = (float*)d_out;

    // 16384 columns, 16 per wave -> 1024 waves -> 256 blocks x 128 threads (4 waves/block).
    dim3 grid(NCOLS / 16 / 4);
    dim3 block(128);
    hipLaunchKernelGGL(biquad_wmma, grid, block, 0, stream, x, cf, qv, gn, out);
}